// _ScaledDotProductAttention_46316927320515
// MI455X (gfx1250) — compile-verified
//
#include <hip/hip_runtime.h>
#include <cstdint>
#include <cstddef>

typedef __attribute__((ext_vector_type(2))) float v2f;
typedef __attribute__((ext_vector_type(4))) float v4f;
typedef __attribute__((ext_vector_type(8))) float v8f;

#define NB      4
#define NH      16
#define S_LEN   2048
#define DK      64
#define ROW_T   16
#define LDS_STRIDE 2052   // 2048 + 4-float pad: half-waves land on disjoint bank groups

__global__ __launch_bounds__(128, 1)
void attn_fp32_wmma_kernel(const float* __restrict__ q,
                           const float* __restrict__ k,      // [B,H,D,S] pre-transposed
                           const float* __restrict__ v,      // [B,H,S,D]
                           const uint8_t* __restrict__ mask, // [B,1,1,S] bool
                           float* __restrict__ ctx_out,      // [B,H,S,D]
                           float* __restrict__ attn_out)     // [B,H,S,S]
{
    __shared__ float p[ROW_T * LDS_STRIDE];   // 128 KB score/prob strip
    __shared__ float redbuf[ROW_T * 8];
    __shared__ float rowmax[ROW_T];
    __shared__ float rowinv[ROW_T];

    const int tid    = threadIdx.x;
    const int wv     = tid >> 5;      // wave id 0..3
    const int ln     = tid & 31;      // lane in wave32
    const int half   = ln >> 4;       // 0 or 1 (half-wave)
    const int lane16 = ln & 15;

    const int bh   = blockIdx.x >> 7;        // (b*H + h); S/16 = 128 row tiles
    const int tile = blockIdx.x & 127;
    const int row0 = tile * ROW_T;
    const int b    = bh >> 4;

    const float*   qbase = q + ((size_t)bh * S_LEN + row0) * DK;
    const float*   kbase = k + (size_t)bh * DK * S_LEN;
    const float*   vbase = v + (size_t)bh * S_LEN * DK;
    const uint8_t* mbase = mask + (size_t)b * S_LEN;
    float*         cbase = ctx_out  + ((size_t)bh * S_LEN + row0) * DK;
    float*         abase = attn_out + ((size_t)bh * S_LEN + row0) * (size_t)S_LEN;

    // ---------------- Phase 1: scores = (Q K^T) * scale, masked -> LDS ----
    // A fragment (16x4 f32): lane = row M, half-wave selects K pair {0,1}/{2,3}
    v2f a[16];
    {
        const float* qrow = qbase + (size_t)lane16 * DK + 2 * half;
        #pragma unroll
        for (int s = 0; s < 16; ++s)
            a[s] = *(const v2f*)(qrow + 4 * s);
    }

    for (int nt = wv; nt < S_LEN / 16; nt += 4) {
        const int n0 = nt * 16;
        const float* kcol = kbase + n0 + lane16;      // k[d][n0+lane16], stride S_LEN per d
        v8f c = {0.f, 0.f, 0.f, 0.f, 0.f, 0.f, 0.f, 0.f};
        #pragma unroll
        for (int s = 0; s < 16; ++s) {
            const int d0 = 4 * s + 2 * half;
            v2f bf;
            bf.x = kcol[(size_t)d0 * S_LEN];
            bf.y = kcol[(size_t)(d0 + 1) * S_LEN];
            c = __builtin_amdgcn_wmma_f32_16x16x4_f32(
                    false, a[s], false, bf, (short)0, c, false, false);
        }
        const bool msk = mbase[n0 + lane16] != 0;     // mask true -> -1e9
        #pragma unroll
        for (int r = 0; r < 8; ++r) {
            const int M = r + 8 * half;               // C/D layout: vgpr r -> rows r, r+8
            float sc = msk ? -1.0e9f : c[r] * 0.125f; // 1/sqrt(64)
            p[M * LDS_STRIDE + n0 + lane16] = sc;
        }
    }
    __syncthreads();

    // ---------------- Phase 2: softmax stats (8 threads per row) ----------
    // Interleaved columns (seg + 8*j): consecutive lanes -> distinct banks.
    const int rrow = tid >> 3;     // 0..15
    const int seg  = tid & 7;      // 0..7
    {
        float mx = -3.0e38f;
        for (int j = 0; j < S_LEN / 8; ++j)
            mx = fmaxf(mx, p[rrow * LDS_STRIDE + seg + 8 * j]);
        redbuf[rrow * 8 + seg] = mx;
    }
    __syncthreads();
    if (tid < 16) {
        float m = redbuf[tid * 8];
        #pragma unroll
        for (int j = 1; j < 8; ++j) m = fmaxf(m, redbuf[tid * 8 + j]);
        rowmax[tid] = m;
    }
    __syncthreads();
    {
        const float rm = rowmax[rrow];
        float sum = 0.f;
        for (int j = 0; j < S_LEN / 8; ++j) {
            const int idx = rrow * LDS_STRIDE + seg + 8 * j;
            float e = __expf(p[idx] - rm);
            p[idx] = e;                       // keep exp() in place
            sum += e;
        }
        redbuf[rrow * 8 + seg] = sum;
    }
    __syncthreads();
    if (tid < 16) {
        float t = 0.f;
        #pragma unroll
        for (int j = 0; j < 8; ++j) t += redbuf[tid * 8 + j];
        rowinv[tid] = 1.0f / t;
    }
    __syncthreads();

    // ------- Phase 3a: stream normalized attn to HBM (b128, non-temporal) --
    for (int idx = tid; idx < ROW_T * (S_LEN / 4); idx += 128) {
        const int r  = idx >> 9;             // 512 float4 per row
        const int cc = (idx & 511) * 4;
        v4f t = *(const v4f*)(p + r * LDS_STRIDE + cc);
        const float inv = rowinv[r];
        t.x *= inv; t.y *= inv; t.z *= inv; t.w *= inv;
        __builtin_nontemporal_store(t, (v4f*)(abase + (size_t)r * S_LEN + cc));
    }

    // ---------------- Phase 3b: context tile = P @ V (K = 2048) -----------
    {
        const int n0 = wv * 16;                       // each wave owns 16 of the 64 cols
        v8f acc0 = {0.f, 0.f, 0.f, 0.f, 0.f, 0.f, 0.f, 0.f};
        v8f acc1 = {0.f, 0.f, 0.f, 0.f, 0.f, 0.f, 0.f, 0.f};
        const float* vcol = vbase + n0 + lane16;      // v[t][n0+lane16], stride DK per t
        const float* prow = p + (size_t)lane16 * LDS_STRIDE + 2 * half;

        for (int k0 = 0; k0 < S_LEN; k0 += 8) {
            v2f a0 = *(const v2f*)(prow + k0);
            v2f b0;
            b0.x = vcol[(size_t)(k0 + 2 * half)     * DK];
            b0.y = vcol[(size_t)(k0 + 2 * half + 1) * DK];
            acc0 = __builtin_amdgcn_wmma_f32_16x16x4_f32(
                       false, a0, false, b0, (short)0, acc0, false, false);

            v2f a1 = *(const v2f*)(prow + k0 + 4);
            v2f b1;
            b1.x = vcol[(size_t)(k0 + 4 + 2 * half)     * DK];
            b1.y = vcol[(size_t)(k0 + 5 + 2 * half)     * DK];
            acc1 = __builtin_amdgcn_wmma_f32_16x16x4_f32(
                       false, a1, false, b1, (short)0, acc1, false, false);
        }

        #pragma unroll
        for (int r = 0; r < 8; ++r) {
            const int M = r + 8 * half;
            float val = (acc0[r] + acc1[r]) * rowinv[M];
            __builtin_nontemporal_store(val, cbase + (size_t)M * DK + n0 + lane16);
        }
    }
}

extern "C" void kernel_launch(void* const* d_in, const int* in_sizes, int n_in,
                              void* d_out, int out_size, void* d_ws, size_t ws_size,
                              hipStream_t stream)
{
    const float*   q    = (const float*)d_in[0];
    const float*   k    = (const float*)d_in[1];
    const float*   v    = (const float*)d_in[2];
    const uint8_t* mask = (const uint8_t*)d_in[3];

    float* ctx  = (float*)d_out;                                   // context first
    float* attn = ctx + (size_t)NB * NH * S_LEN * DK;              // then attn

    const int grid = NB * NH * (S_LEN / ROW_T);                    // 8192 workgroups
    attn_fp32_wmma_kernel<<<grid, 128, 0, stream>>>(q, k, v, mask, ctx, attn);
}